// DGLMHA_46651934769542
// MI455X (gfx1250) — compile-verified
//
#include <hip/hip_runtime.h>
#include <math.h>

typedef __attribute__((ext_vector_type(2))) float v2f;
typedef __attribute__((ext_vector_type(8))) float v8f;

#define HID 256            // hidden size
#define NHEAD 8
#define DK 32
#define LDA 260            // padded LDS stride (260*4B -> bank = r*4, conflict-free)

// ---------------------------------------------------------------------------
// C[M,HID] = A[M,HID] @ W[HID,HID] + bias, fp32 WMMA 16x16x4.
// One block = 16 output rows; 8 waves each compute two 16x16 N-tiles.
// bias must be a valid pointer (use a zeroed scratch region for "no bias").
// ---------------------------------------------------------------------------
__global__ __launch_bounds__(256) void gemm_bias_f32(
    const float* __restrict__ A, const float* __restrict__ W,
    const float* __restrict__ bias, float* __restrict__ C, int M) {
  __shared__ float As[16 * LDA];

  const int m0 = blockIdx.x * 16;
  const bool full_tile = (m0 + 16 <= M);

  // Cooperative load of the 16x256 A slab into padded LDS (float4 per thread x4).
  for (int i = threadIdx.x; i < 16 * 64; i += 256) {
    int r  = i >> 6;          // row 0..15
    int c4 = (i & 63) * 4;    // col 0..252 step 4
    int row = m0 + r;
    if (row >= M) row = M - 1;                 // clamp (harmless duplicate read)
    float4 t = *(const float4*)(A + (size_t)row * HID + c4);
    *(float4*)(&As[r * LDA + c4]) = t;
  }
  __syncthreads();

  const int wave = threadIdx.x >> 5;
  const int lane = threadIdx.x & 31;
  const int r    = lane & 15;          // row within tile (A) / col within tile (B,C)
  const int kk   = (lane >> 4) * 2;    // K sub-offset: lanes 0-15 -> K0,K1 ; 16-31 -> K2,K3

  for (int nt = 0; nt < 2; ++nt) {
    const int n0 = (wave * 2 + nt) * 16;
    v8f acc = {};
#pragma unroll 8
    for (int k0 = 0; k0 < HID; k0 += 4) {
      v2f a, b;
      // A 16x4 fragment: lane r holds row m0+r; VGPR0/1 = K = k0+kk, k0+kk+1
      a.x = As[r * LDA + k0 + kk];
      a.y = As[r * LDA + k0 + kk + 1];
      // B 4x16 fragment: lane r holds col n0+r; VGPR0/1 = K rows k0+kk, k0+kk+1
      b.x = W[(size_t)(k0 + kk) * HID + n0 + r];
      b.y = W[(size_t)(k0 + kk + 1) * HID + n0 + r];
      acc = __builtin_amdgcn_wmma_f32_16x16x4_f32(
          /*neg_a=*/false, a, /*neg_b=*/false, b,
          /*c_mod=*/(short)0, acc, /*reuse_a=*/false, /*reuse_b=*/false);
    }
    // C/D layout: VGPR i -> M = m0 + (lane>>4)*8 + i, N = n0 + (lane&15)
    const int n     = n0 + r;
    const int mbase = m0 + (lane >> 4) * 8;
    const float bv  = bias[n];
    if (full_tile) {
#pragma unroll
      for (int i = 0; i < 8; ++i)
        C[(size_t)(mbase + i) * HID + n] = acc[i] + bv;
    } else {
#pragma unroll
      for (int i = 0; i < 8; ++i) {
        int m = mbase + i;
        if (m < M) C[(size_t)m * HID + n] = acc[i] + bv;
      }
    }
  }
}

// ---------------------------------------------------------------------------
// One wave per edge: score[h] = exp(clip(dot(k[src],q[dst])_h / sqrt(dk),-5,5))
// then atomically scatter v[src]*score into WV[dst] and score into Z[dst].
// Lane l owns 8 contiguous dims [l*8, l*8+8) -> entirely inside head l>>2.
// Non-returning f32 atomics -> cheap L2 RMW path.
// ---------------------------------------------------------------------------
__global__ __launch_bounds__(256) void edge_kernel(
    const float* __restrict__ Q, const float* __restrict__ K,
    const float* __restrict__ V, const int* __restrict__ src,
    const int* __restrict__ dst, float* __restrict__ WV,
    float* __restrict__ Z, int E) {
  const int wave = threadIdx.x >> 5;
  const int lane = threadIdx.x & 31;
  const int e = blockIdx.x * 8 + wave;
  if (e >= E) return;

  const int s = src[e];
  const int d = dst[e];

  const float* kr = K + (size_t)s * HID + lane * 8;
  const float* qr = Q + (size_t)d * HID + lane * 8;
  float4 k0 = *(const float4*)(kr);
  float4 k1 = *(const float4*)(kr + 4);
  float4 q0 = *(const float4*)(qr);
  float4 q1 = *(const float4*)(qr + 4);

  float p = k0.x * q0.x + k0.y * q0.y + k0.z * q0.z + k0.w * q0.w +
            k1.x * q1.x + k1.y * q1.y + k1.z * q1.z + k1.w * q1.w;
  // butterfly reduce within each 4-lane head group -> all 4 lanes hold head sum
  p += __shfl_xor(p, 1, 32);
  p += __shfl_xor(p, 2, 32);

  const float inv_scale = 0.17677669529663687f;  // 1/sqrt(32)
  float sc = p * inv_scale;
  sc = fminf(fmaxf(sc, -5.0f), 5.0f);
  sc = __expf(sc);

  const float* vr = V + (size_t)s * HID + lane * 8;
  float4 v0 = *(const float4*)(vr);
  float4 v1 = *(const float4*)(vr + 4);

  float* wvp = WV + (size_t)d * HID + lane * 8;
  atomicAdd(wvp + 0, v0.x * sc);
  atomicAdd(wvp + 1, v0.y * sc);
  atomicAdd(wvp + 2, v0.z * sc);
  atomicAdd(wvp + 3, v0.w * sc);
  atomicAdd(wvp + 4, v1.x * sc);
  atomicAdd(wvp + 5, v1.y * sc);
  atomicAdd(wvp + 6, v1.z * sc);
  atomicAdd(wvp + 7, v1.w * sc);

  if ((lane & 3) == 0) atomicAdd(Z + (size_t)d * NHEAD + (lane >> 2), sc);
}

// ---------------------------------------------------------------------------
// WV[n, h, :] /= Z[n, h]   (in place)
// ---------------------------------------------------------------------------
__global__ __launch_bounds__(256) void normalize_kernel(
    float* __restrict__ WV, const float* __restrict__ Z, int Ns) {
  size_t idx = (size_t)blockIdx.x * 256 + threadIdx.x;
  if (idx >= (size_t)Ns * HID) return;
  int n = (int)(idx >> 8);
  int c = (int)(idx & (HID - 1));
  WV[idx] = WV[idx] / Z[(size_t)n * NHEAD + (c >> 5)];
}

// ---------------------------------------------------------------------------
extern "C" void kernel_launch(void* const* d_in, const int* in_sizes, int n_in,
                              void* d_out, int out_size, void* d_ws, size_t ws_size,
                              hipStream_t stream) {
  const float* context = (const float*)d_in[0];  // [Nq, 256]
  const float* node    = (const float*)d_in[1];  // [Ns, 256]
  const float* Wq      = (const float*)d_in[2];
  const float* bq      = (const float*)d_in[3];
  const float* Wk      = (const float*)d_in[4];
  const float* Wv      = (const float*)d_in[5];
  const float* Wo      = (const float*)d_in[6];
  const float* bo      = (const float*)d_in[7];
  const int*   src     = (const int*)d_in[8];
  const int*   dst     = (const int*)d_in[9];

  const int Nq = in_sizes[0] / HID;
  const int Ns = in_sizes[1] / HID;
  const int E  = in_sizes[8];

  float* ws   = (float*)d_ws;
  float* Q    = ws;                          // [Ns, 256]  (queries come from `node`)
  float* K    = Q  + (size_t)Ns * HID;       // [Nq, 256]
  float* V    = K  + (size_t)Nq * HID;       // [Nq, 256]
  float* WV   = V  + (size_t)Nq * HID;       // [Ns, 256]
  float* Z    = WV + (size_t)Ns * HID;       // [Ns, 8]
  float* zero = Z  + (size_t)Ns * NHEAD;     // [256] zero bias scratch

  // zero accumulators + zero-bias scratch in one memset (all adjacent)
  hipMemsetAsync(WV, 0,
                 ((size_t)Ns * HID + (size_t)Ns * NHEAD + HID) * sizeof(float),
                 stream);

  dim3 blk(256);
  // projections
  gemm_bias_f32<<<(Ns + 15) / 16, blk, 0, stream>>>(node,    Wq, bq,   Q, Ns);
  gemm_bias_f32<<<(Nq + 15) / 16, blk, 0, stream>>>(context, Wk, zero, K, Nq);
  gemm_bias_f32<<<(Nq + 15) / 16, blk, 0, stream>>>(context, Wv, zero, V, Nq);

  // edge scatter (L2-resident gather/scatter; 8 waves = 8 edges per block)
  edge_kernel<<<(E + 7) / 8, blk, 0, stream>>>(Q, K, V, src, dst, WV, Z, E);

  // normalize
  int tot = Ns * HID;
  normalize_kernel<<<(tot + 255) / 256, blk, 0, stream>>>(WV, Z, Ns);

  // output projection
  gemm_bias_f32<<<(Ns + 15) / 16, blk, 0, stream>>>(WV, Wo, bo, (float*)d_out, Ns);
}